// ReversiWasmlModel_47072841564550
// MI455X (gfx1250) — compile-verified
//
#include <hip/hip_runtime.h>

typedef float v2f __attribute__((ext_vector_type(2)));
typedef float v4f __attribute__((ext_vector_type(4)));
typedef float v8f __attribute__((ext_vector_type(8)));

#define LINPUT   256
#define KFEAT    32
#define NBUCKET  60
#define TILE_M   16
#define XSTRIDE  260   // padded LDS row stride (floats): 260%64=4 -> conflict-free strided reads

__global__ __launch_bounds__(256)
void nnue_fused_kernel(const int* __restrict__ feat,
                       const int* __restrict__ ply,
                       const float* __restrict__ emb_w,
                       const float* __restrict__ emb_b,
                       const float* __restrict__ out_w,
                       const float* __restrict__ out_b,
                       float* __restrict__ out,
                       int B)
{
    __shared__ int   s_idx[TILE_M * KFEAT];     // 2 KB: staged feature indices
    __shared__ float s_x[TILE_M * XSTRIDE];     // 16.25 KB: activated accumulator tile
    __shared__ float s_out[TILE_M * 64];        // 4 KB: 16x64 WMMA result (60 valid cols)

    const int tid = threadIdx.x;
    const int b0  = blockIdx.x * TILE_M;

    // ---------------- stage the 16x32 indices into LDS ----------------
    {
        const int gmax = B * KFEAT - 1;
        int g = b0 * KFEAT + tid;
        s_idx[tid]       = feat[min(g, gmax)];
        s_idx[tid + 256] = feat[min(g + 256, gmax)];
    }
    __syncthreads();

    // ---------------- gather + accumulate (memory-bound core) ----------------
    // thread (s,c): sample s = tid>>4, chunk c = tid&15 -> dims [16c, 16c+16)
    {
        const int s  = tid >> 4;
        const int c  = tid & 15;
        const int d0 = c * 16;

        v4f acc0 = *(const v4f*)(emb_b + d0 + 0);
        v4f acc1 = *(const v4f*)(emb_b + d0 + 4);
        v4f acc2 = *(const v4f*)(emb_b + d0 + 8);
        v4f acc3 = *(const v4f*)(emb_b + d0 + 12);

        #pragma unroll
        for (int k = 0; k < KFEAT; ++k) {
            const int idx = s_idx[s * KFEAT + k];
            const float* row = emb_w + (size_t)idx * LINPUT + d0;
            if (k + 6 < KFEAT) {   // compile-time after full unroll -> branch-free prefetch
                const int pidx = s_idx[s * KFEAT + k + 6];
                __builtin_prefetch(emb_w + (size_t)pidx * LINPUT + d0, 0, 1);
            }
            acc0 += *(const v4f*)(row + 0);
            acc1 += *(const v4f*)(row + 4);
            acc2 += *(const v4f*)(row + 8);
            acc3 += *(const v4f*)(row + 12);
        }

        // clamp-square activation with quantization scale
        v4f accs[4] = {acc0, acc1, acc2, acc3};
        float* dst = s_x + s * XSTRIDE + d0;
        #pragma unroll
        for (int j = 0; j < 4; ++j) {
            v4f v = accs[j];
            #pragma unroll
            for (int e = 0; e < 4; ++e) {
                float t = fminf(fmaxf(v[e], 0.0f), 1.0f);
                v[e] = t * t * (255.0f / 256.0f);
            }
            *(v4f*)(dst + 4 * j) = v;
        }
    }
    __syncthreads();

    // ---------------- 16x256 @ 256x64 dense via V_WMMA_F32_16X16X4_F32 ----------------
    // Waves 0..3 each compute one 16-wide column tile (60 cols padded to 64).
    // Columns 60..63 compute garbage from a clamped (in-bounds, finite) out_w row;
    // they are never read back, so NO zero-masking is needed -> branch-free B loads.
    const int wave = tid >> 5;
    const int lane = tid & 31;
    if (wave < 4) {
        const int nTile  = wave;
        const int half   = lane >> 4;      // 0: lanes 0-15, 1: lanes 16-31
        const int lane16 = lane & 15;
        const int ncol   = min(nTile * 16 + lane16, NBUCKET - 1);  // clamp: stay in-bounds
        const float* brow = out_w + (size_t)ncol * LINPUT;
        const float* arow = s_x + lane16 * XSTRIDE;

        v8f cacc = {};
        #pragma unroll 8
        for (int kk = 0; kk < LINPUT; kk += 4) {
            // ISA layout: lanes 0-15 carry K = kk+0/kk+1, lanes 16-31 carry K = kk+2/kk+3
            const int kA = kk + 2 * half;
            v2f a, b;
            a.x = arow[kA];        // ds_load_b64
            a.y = arow[kA + 1];
            b.x = brow[kA];        // global_load_b64, unconditional, L2-resident
            b.y = brow[kA + 1];
            cacc = __builtin_amdgcn_wmma_f32_16x16x4_f32(
                /*neg_a=*/false, a, /*neg_b=*/false, b,
                /*c_mod=*/(short)0, cacc, /*reuse_a=*/false, /*reuse_b=*/false);
        }

        // D layout: VGPR r, lanes 0-15 -> M=r, lanes 16-31 -> M=r+8; N = lane16 within tile
        #pragma unroll
        for (int r = 0; r < 8; ++r) {
            const int m = r + 8 * half;
            s_out[m * 64 + nTile * 16 + lane16] = cacc[r];
        }
    }
    __syncthreads();

    // ---------------- per-sample bucket select + bias ----------------
    if (tid < TILE_M) {
        const int b = b0 + tid;
        if (b < B) {
            int bucket = ply[b];
            bucket = bucket < 0 ? 0 : (bucket >= NBUCKET ? NBUCKET - 1 : bucket);
            out[b] = s_out[tid * 64 + bucket] + out_b[bucket];
        }
    }
}

extern "C" void kernel_launch(void* const* d_in, const int* in_sizes, int n_in,
                              void* d_out, int out_size, void* d_ws, size_t ws_size,
                              hipStream_t stream) {
    const int*   feat  = (const int*)  d_in[0];   // feature_indices [B,32]
    // d_in[1] = mobility [B,1] -- not used by the reference computation
    const int*   ply   = (const int*)  d_in[2];   // ply [B]
    const float* emb_w = (const float*)d_in[3];   // [500000,256]
    const float* emb_b = (const float*)d_in[4];   // [256]
    const float* out_w = (const float*)d_in[5];   // [60,256]
    const float* out_b = (const float*)d_in[6];   // [60]
    float* out = (float*)d_out;                   // [B,1]

    const int B = in_sizes[0] / KFEAT;
    const int grid = (B + TILE_M - 1) / TILE_M;
    nnue_fused_kernel<<<grid, 256, 0, stream>>>(feat, ply, emb_w, emb_b, out_w, out_b, out, B);

    (void)n_in; (void)out_size; (void)d_ws; (void)ws_size;
}